// MRLELoss_24850680775337
// MI455X (gfx1250) — compile-verified
//
#include <hip/hip_runtime.h>
#include <hip/hip_bf16.h>

typedef __attribute__((ext_vector_type(2))) float v2f;
typedef __attribute__((ext_vector_type(8))) float v8f;

__global__ void MRLELoss_zero_kernel(float* out) { out[0] = 0.0f; }

// One block per row i. 256 threads = 8 wave32s. Each wave owns 128 classes
// (8 tiles of 16); class space padded to 1024 with base logit = -1e30.
// D processed in chunks of 16 steps; per chunk each tile's 16-step prefix
// deltas are computed as a triangular 16x16x16 matmul via 4 chained
// V_WMMA_F32_16X16X4_F32 (full fp32 precision, matches reference math).
__global__ __launch_bounds__(256) void MRLELoss_24850680775337_kernel(
    const float* __restrict__ Z,      // [B, D]
    const int*   __restrict__ labels, // [B]
    const float* __restrict__ W,      // [C, D]
    float* __restrict__ out,          // [1]
    int B, int D, int C)
{
    const int row  = blockIdx.x;
    const int tid  = threadIdx.x;
    const int wave = tid >> 5;
    const int lane = tid & 31;
    const int lr   = lane & 15;   // 0..15
    const int hi   = lane >> 4;   // 0 or 1 (lane half)

    __shared__ float lds_z[16];       // z chunk broadcast
    __shared__ float lds_p[16];       // z[d]*W[y,d] terms for label prefix
    __shared__ float lds_max[16][8];  // per-step per-wave max
    __shared__ float lds_sum[16][8];  // per-step per-wave sum(exp)
    __shared__ float lds_gmax[16];    // per-step global max
    __shared__ float lds_red[16];     // final loss reduction

    const int y = min(max(labels[row], 0), C - 1);

    // Per-tile class (clamped for safe addressing) and running base logit.
    int   cls[8];
    float base[8];
#pragma unroll
    for (int t = 0; t < 8; ++t) {
        int c   = wave * 128 + t * 16 + lr;
        cls[t]  = (c < C) ? c : (C - 1);
        base[t] = (c < C) ? 0.0f : -1e30f;  // pad classes never contribute
    }

    float labbase = 0.0f;   // prefix dot(z, W[y]) (maintained by tid<16)
    float lacc    = 0.0f;   // per-step loss accumulator (tid<16: step j=tid)

    const int nchunk = D >> 4;
    for (int ch = 0; ch < nchunk; ++ch) {
        const int k0 = ch << 4;

        if (tid < 16) lds_z[tid] = Z[(size_t)row * D + k0 + tid];
        __syncthreads();                                            // B1
        if (tid < 16) lds_p[tid] = lds_z[tid] * W[(size_t)y * D + k0 + tid];

        // A fragments: A[j,d] = (d<=j) ? z[k0+d] : 0, split into 4 K=4 slabs.
        // Layout (16x4 f32 A): VGPR0 holds K=2*hi, VGPR1 holds K=2*hi+1.
        v2f aq[4];
#pragma unroll
        for (int q = 0; q < 4; ++q) {
            int d0 = 4 * q + 2 * hi;
            aq[q].x = (d0     <= lr) ? lds_z[d0]     : 0.0f;
            aq[q].y = (d0 + 1 <= lr) ? lds_z[d0 + 1] : 0.0f;
        }

        // Per-thread logits for this chunk: [tile][v], step j = v + 8*hi,
        // class = cls[tile] (column = lane%16 of the tile).
        float lg[8][8];
#pragma unroll
        for (int t = 0; t < 8; ++t) {
            v8f acc = {0.f, 0.f, 0.f, 0.f, 0.f, 0.f, 0.f, 0.f};
            // B fragment (4x16): lanes 0-15 carry K0/K1, lanes 16-31 K2/K3
            // -> two consecutive floats of W row `cls[t]` at k0+4q+2*hi.
            const float* wp = W + (size_t)cls[t] * D + k0 + 2 * hi;
#pragma unroll
            for (int q = 0; q < 4; ++q) {
                v2f b;
                b.x = wp[4 * q];
                b.y = wp[4 * q + 1];
                acc = __builtin_amdgcn_wmma_f32_16x16x4_f32(
                    false, aq[q], false, b, (short)0, acc, false, false);
            }
#pragma unroll
            for (int v = 0; v < 8; ++v) lg[t][v] = base[t] + acc[v];
            // Advance base by the full-chunk delta (row j=15 lives in v=7 of
            // the upper lane half); duplicate to both halves via shuffle.
            float last = __shfl(acc[7], lr + 16, 32);
            base[t] += last;
        }

        // ---- Pass 1: per-step max over all 1024 classes ----
        float m[8];
#pragma unroll
        for (int v = 0; v < 8; ++v) {
            float mv = lg[0][v];
#pragma unroll
            for (int t = 1; t < 8; ++t) mv = fmaxf(mv, lg[t][v]);
#pragma unroll
            for (int s = 1; s < 16; s <<= 1)
                mv = fmaxf(mv, __shfl_xor(mv, s, 32));
            m[v] = mv;
        }
        if (lr == 0) {
#pragma unroll
            for (int v = 0; v < 8; ++v) lds_max[v + 8 * hi][wave] = m[v];
        }
        __syncthreads();                                            // B2
        if (tid < 16) {
            float g = lds_max[tid][0];
#pragma unroll
            for (int w = 1; w < 8; ++w) g = fmaxf(g, lds_max[tid][w]);
            lds_gmax[tid] = g;
        }
        __syncthreads();                                            // B3

        // ---- Pass 2: per-step sum of exp(logit - gmax) ----
#pragma unroll
        for (int v = 0; v < 8; ++v) {
            float gm = lds_gmax[v + 8 * hi];
            float sv = 0.0f;
#pragma unroll
            for (int t = 0; t < 8; ++t) sv += __expf(lg[t][v] - gm);
#pragma unroll
            for (int s = 1; s < 16; s <<= 1) sv += __shfl_xor(sv, s, 32);
            if (lr == 0) lds_sum[v + 8 * hi][wave] = sv;
        }
        __syncthreads();                                            // B4

        // ---- Loss for the 16 steps of this chunk (thread j = tid) ----
        if (tid < 16) {
            float S = 0.0f;
#pragma unroll
            for (int w = 0; w < 8; ++w) S += lds_sum[tid][w];
            float lse = lds_gmax[tid] + __logf(S);
            float lab = labbase;
            for (int d = 0; d <= tid; ++d) lab += lds_p[d];
            lacc += lse - lab;
            float tot = lab;
            for (int d = tid + 1; d < 16; ++d) tot += lds_p[d];
            labbase = tot;  // = old labbase + full chunk sum
        }
        // Next chunk's first shared write happens after B1, so no extra
        // barrier is needed here (B2..B4 already separate all RAW/WAR pairs).
    }

    if (tid < 16) lds_red[tid] = lacc;
    __syncthreads();
    if (tid == 0) {
        float tot = 0.0f;
#pragma unroll
        for (int t = 0; t < 16; ++t) tot += lds_red[t];
        atomicAdd(out, tot / ((float)B * (float)D));
    }
}

extern "C" void kernel_launch(void* const* d_in, const int* in_sizes, int n_in,
                              void* d_out, int out_size, void* d_ws, size_t ws_size,
                              hipStream_t stream) {
    const float* Z      = (const float*)d_in[0];
    const int*   labels = (const int*)d_in[1];
    const float* W      = (const float*)d_in[2];
    float*       out    = (float*)d_out;

    const int B = in_sizes[1];           // 512
    const int D = in_sizes[0] / B;       // 256
    const int C = in_sizes[2] / D;       // 1000

    MRLELoss_zero_kernel<<<1, 1, 0, stream>>>(out);
    MRLELoss_24850680775337_kernel<<<B, 256, 0, stream>>>(Z, labels, W, out, B, D, C);
}